// HungarianLossComputation_70712341562216
// MI455X (gfx1250) — compile-verified
//
#include <hip/hip_runtime.h>
#include <hip/hip_bf16.h>

typedef __attribute__((ext_vector_type(2))) float v2f;
typedef __attribute__((ext_vector_type(8))) float v8f;

static constexpr int kB = 32;
static constexpr int kT = 2048;
static constexpr int kO = 128;
static constexpr int kC = 80;
static constexpr float kEPS = 1e-7f;

// ---------------------------------------------------------------------------
// Kernel 0: logsumexp over classes, one thread per (b,t). 20x float4 loads.
// ---------------------------------------------------------------------------
__global__ void lse_kernel(const float* __restrict__ pred_classes,
                           float* __restrict__ lse) {
    int idx = blockIdx.x * blockDim.x + threadIdx.x;   // [0, B*T)
    if (idx >= kB * kT) return;
    const float4* row = (const float4*)(pred_classes + (size_t)idx * kC);
    float m = -__builtin_inff();
    float4 v[20];
#pragma unroll
    for (int q = 0; q < kC / 4; ++q) {
        v[q] = row[q];
        m = fmaxf(m, fmaxf(fmaxf(v[q].x, v[q].y), fmaxf(v[q].z, v[q].w)));
    }
    float s = 0.f;
#pragma unroll
    for (int q = 0; q < kC / 4; ++q) {
        s += __expf(v[q].x - m) + __expf(v[q].y - m) +
             __expf(v[q].z - m) + __expf(v[q].w - m);
    }
    lse[idx] = m + __logf(s);
}

// ---------------------------------------------------------------------------
// Kernel 1: cost tiles. Grid = B*(T/16) blocks, 256 threads = 8 waves.
// Wave w computes the 16x16 tile (rows t0..t0+15, cols n0=w*16..+15).
// Class gather done as fp32 WMMA against a one-hot B matrix (bit exact).
// Output stored TRANSPOSED: costT[b][o][t].
// ---------------------------------------------------------------------------
__global__ void cost_tile_kernel(const float* __restrict__ pred_bboxes,
                                 const float* __restrict__ target_bboxes,
                                 const float* __restrict__ pred_classes,
                                 const int*   __restrict__ target_classes,
                                 const float* __restrict__ lse,
                                 float* __restrict__ costT) {
    const int tilesPerBatch = kT / 16;
    int b  = blockIdx.x / tilesPerBatch;
    int t0 = (blockIdx.x % tilesPerBatch) * 16;
    int wave = threadIdx.x >> 5;        // 0..7
    int lane = threadIdx.x & 31;
    int n0 = wave * 16;
    int lo = lane & 15;                 // row (A) / col (B,C) index
    int hi = lane >> 4;                 // 0 or 1

    // B operand one-hot column: target class for column n0+lo
    int tcn = target_classes[b * kO + n0 + lo];

    // A operand rows: logits row (t0 + lo); per K-chunk this lane needs
    // elements kbase + 2*hi + {0,1} (contiguous -> float2 load, 8B aligned).
    const float* arow = pred_classes + ((size_t)b * kT + t0 + lo) * kC;

    v8f acc = {};
#pragma unroll
    for (int k = 0; k < kC; k += 4) {
        v2f a = *(const v2f*)(arow + k + 2 * hi);
        v2f bb;
        bb.x = (tcn == k + 2 * hi + 0) ? 1.0f : 0.0f;
        bb.y = (tcn == k + 2 * hi + 1) ? 1.0f : 0.0f;
        acc = __builtin_amdgcn_wmma_f32_16x16x4_f32(
            /*neg_a=*/false, a, /*neg_b=*/false, bb,
            /*c_mod=*/(short)0, acc, /*reuse_a=*/false, /*reuse_b=*/false);
    }
    // acc[r] = logits[b, t0 + r + 8*hi, tc[b, n0+lo]]

    // Target box for this lane's column n (shared across the 8 rows).
    int n = n0 + lo;
    float4 tb = ((const float4*)target_bboxes)[b * kO + n];
    float b2x1 = tb.x - tb.z * 0.5f, b2y1 = tb.y - tb.w * 0.5f;
    float b2x2 = tb.x + tb.z * 0.5f, b2y2 = tb.y + tb.w * 0.5f;
    float w2 = b2x2 - b2x1, h2 = b2y2 - b2y1 + kEPS;

#pragma unroll
    for (int r = 0; r < 8; ++r) {
        int m = r + 8 * hi;
        int t = t0 + m;
        float4 pb = ((const float4*)pred_bboxes)[(size_t)b * kT + t];
        float b1x1 = pb.x - pb.z * 0.5f, b1y1 = pb.y - pb.w * 0.5f;
        float b1x2 = pb.x + pb.z * 0.5f, b1y2 = pb.y + pb.w * 0.5f;
        float w1 = b1x2 - b1x1, h1 = b1y2 - b1y1 + kEPS;
        float iw = fmaxf(fminf(b1x2, b2x2) - fmaxf(b1x1, b2x1), 0.f);
        float ih = fmaxf(fminf(b1y2, b2y2) - fmaxf(b1y1, b2y1), 0.f);
        float inter = iw * ih;
        float uni = w1 * h1 + w2 * h2 - inter + kEPS;
        float iou = inter / uni;
        float cw = fmaxf(b1x2, b2x2) - fminf(b1x1, b2x1);
        float ch = fmaxf(b1y2, b2y2) - fminf(b1y1, b2y1);
        float carea = cw * ch + kEPS;
        float giou = iou - (carea - uni) / carea;
        float gl = 1.0f - giou;                         // giou loss
        float cls = lse[b * kT + t] - acc[r];           // -log_softmax gather
        costT[((size_t)b * kO + n) * kT + t] = gl + cls;
    }
}

// ---------------------------------------------------------------------------
// Kernel 2: Jonker-Volgenant assignment, one block per batch (32 blocks),
// 256 threads. All state in LDS (~37KB << 320KB/WGP). Cost rows (8KB each)
// are re-read from global per path step; whole 32MB matrix is L2 resident.
// ---------------------------------------------------------------------------
__global__ void hungarian_kernel(const float* __restrict__ costT,
                                 int* __restrict__ pred_idx,
                                 int* __restrict__ tgt_idx) {
    __shared__ float sh_shortest[kT];
    __shared__ float sh_v[kT];
    __shared__ int   sh_path[kT];
    __shared__ int   sh_row4col[kT];
    __shared__ unsigned char sh_SC[kT];
    __shared__ float sh_u[kO];
    __shared__ unsigned char sh_SR[kO];
    __shared__ int   sh_col4row[kO];
    __shared__ float red_val[256];
    __shared__ int   red_idx[256];
    __shared__ int   s_i, s_sink;
    __shared__ float s_minval;

    const int b = blockIdx.x;
    const float* c = costT + (size_t)b * kO * kT;
    const int tid = threadIdx.x;
    const int nt  = blockDim.x;   // 256
    const float INF = __builtin_inff();

    for (int j = tid; j < kT; j += nt) { sh_v[j] = 0.f; sh_row4col[j] = -1; }
    for (int i = tid; i < kO; i += nt) { sh_u[i] = 0.f; sh_col4row[i] = -1; }
    __syncthreads();

    for (int cur = 0; cur < kO; ++cur) {
        for (int j = tid; j < kT; j += nt) {
            sh_shortest[j] = INF; sh_path[j] = -1; sh_SC[j] = 0;
        }
        for (int i = tid; i < kO; i += nt) sh_SR[i] = 0;
        if (tid == 0) { s_i = cur; s_minval = 0.f; s_sink = -1; }
        __syncthreads();

        while (true) {
            int   i  = s_i;
            float mv = s_minval;
            if (tid == 0) sh_SR[i] = 1;
            float ui = sh_u[i];
            const float* ci = c + (size_t)i * kT;
            // relax: shortest[j] = min(shortest[j], mv + c[i][j] - u[i] - v[j])
            for (int j = tid; j < kT; j += nt) {
                if (!sh_SC[j]) {
                    float r = mv + ci[j] - ui - sh_v[j];
                    if (r < sh_shortest[j]) { sh_shortest[j] = r; sh_path[j] = i; }
                }
            }
            __syncthreads();
            // block argmin over masked shortest (first-min tie-break like numpy)
            float best = INF; int bj = kT;
            for (int j = tid; j < kT; j += nt) {
                float val = sh_SC[j] ? INF : sh_shortest[j];
                if (val < best) { best = val; bj = j; }
            }
            red_val[tid] = best; red_idx[tid] = bj;
            __syncthreads();
            for (int s = nt >> 1; s > 0; s >>= 1) {
                if (tid < s) {
                    float ov = red_val[tid + s]; int oj = red_idx[tid + s];
                    if (ov < red_val[tid] ||
                        (ov == red_val[tid] && oj < red_idx[tid])) {
                        red_val[tid] = ov; red_idx[tid] = oj;
                    }
                }
                __syncthreads();
            }
            if (tid == 0) {
                int j = red_idx[0];
                s_minval = red_val[0];
                sh_SC[j] = 1;
                if (sh_row4col[j] == -1) s_sink = j;
                else                     s_i = sh_row4col[j];
            }
            __syncthreads();
            if (s_sink != -1) break;
        }

        // dual updates
        float mv = s_minval;
        int sink = s_sink;
        if (tid == 0) sh_u[cur] += mv;
        for (int r = tid; r < kO; r += nt) {
            if (r != cur && sh_SR[r])
                sh_u[r] += mv - sh_shortest[sh_col4row[r]];
        }
        for (int j = tid; j < kT; j += nt) {
            if (sh_SC[j]) sh_v[j] -= mv - sh_shortest[j];
        }
        __syncthreads();

        // augment along path (serial, thread 0)
        if (tid == 0) {
            int j = sink;
            while (true) {
                int i = sh_path[j];
                sh_row4col[j] = i;
                int tmp = sh_col4row[i];
                sh_col4row[i] = j;
                j = tmp;
                if (i == cur) break;
            }
        }
        __syncthreads();
    }

    // order = argsort(col4row) via rank counting (all values distinct)
    for (int i = tid; i < kO; i += nt) {
        int my = sh_col4row[i];
        int rank = 0;
        for (int k = 0; k < kO; ++k) rank += (sh_col4row[k] < my) ? 1 : 0;
        pred_idx[b * kO + rank] = my;   // token index
        tgt_idx[b * kO + rank]  = i;    // target index
    }
}

// ---------------------------------------------------------------------------
extern "C" void kernel_launch(void* const* d_in, const int* in_sizes, int n_in,
                              void* d_out, int out_size, void* d_ws, size_t ws_size,
                              hipStream_t stream) {
    (void)in_sizes; (void)n_in; (void)out_size; (void)ws_size;
    const float* pred_bboxes    = (const float*)d_in[0];
    const float* target_bboxes  = (const float*)d_in[1];
    const float* pred_classes   = (const float*)d_in[2];
    const int*   target_classes = (const int*)d_in[3];

    float* costT = (float*)d_ws;                                   // 32 MB
    float* lse   = (float*)((char*)d_ws +
                            (size_t)kB * kO * kT * sizeof(float)); // +256 KB

    int* out      = (int*)d_out;
    int* pred_idx = out;                 // [B,O] int32
    int* tgt_idx  = out + kB * kO;       // [B,O] int32

    lse_kernel<<<(kB * kT + 255) / 256, 256, 0, stream>>>(pred_classes, lse);
    cost_tile_kernel<<<kB * (kT / 16), 256, 0, stream>>>(
        pred_bboxes, target_bboxes, pred_classes, target_classes, lse, costT);
    hungarian_kernel<<<kB, 256, 0, stream>>>(costT, pred_idx, tgt_idx);
}